// PatchWiseDeconv2d_83854941487371
// MI455X (gfx1250) — compile-verified
//
#include <hip/hip_runtime.h>
#include <hip/hip_bf16.h>

// PatchWiseDeconv2d on gfx1250:
//   per-position GEMM (M=batch=16, K=cin=64, N=cout*16taps) via
//   V_WMMA_F32_16X16X4_F32, overlap-add fold via 4 parity phases (no atomics).

typedef __attribute__((ext_vector_type(2))) float v2f;
typedef __attribute__((ext_vector_type(8))) float v8f;

#define B_    16
#define CIN_  64
#define COUT_ 64
#define HW_   32
#define POS_  (HW_ * HW_)        // 1024
#define TAPS_ 16
#define HOUT_ 66
#define LDA_  68                 // padded LDS row stride (bank-conflict free)

__global__ __launch_bounds__(256)
void bias_broadcast_kernel(const float* __restrict__ bias,
                           float* __restrict__ out, int total, int per) {
    int idx = blockIdx.x * blockDim.x + threadIdx.x;
    if (idx < total) out[idx] = bias[idx % per];
}

// One block per position in this parity class (256 blocks). 8 waves; each
// wave computes 8 cout tiles of D(16x16) = A(16x64) x B(64x16), K split into
// 16 WMMA_F32_16X16X4_F32 steps, then overlap-adds into out (unique writer
// per location within a phase).
__global__ __launch_bounds__(256)
void deconv_wmma_kernel(const float* __restrict__ x,
                        const float* __restrict__ w,
                        float* __restrict__ out, int par) {
    __shared__ float Alds[B_ * LDA_];

    const int blk = blockIdx.x;                       // 0..255
    const int ph  = (((blk >> 4) & 15) << 1) | (par >> 1);
    const int pw  = ((blk & 15) << 1) | (par & 1);
    const int p   = ph * HW_ + pw;                    // position index

    // ---- stage x[:, :, ph, pw] (16 x 64 fp32) into LDS ----
    for (int e = threadIdx.x; e < B_ * CIN_; e += 256) {
        int b = e >> 6, i = e & 63;
        Alds[b * LDA_ + i] = x[(size_t)(b * CIN_ + i) * POS_ + p];
    }
    __syncthreads();

    const int lane  = threadIdx.x & 31;
    const int wv    = threadIdx.x >> 5;               // wave 0..7
    const int row   = lane & 15;                      // M (A) / N=tap (B,D)
    const int khalf = (lane >> 4) << 1;               // 0 (lanes 0-15) or 2

    // ---- preload A in WMMA 16x4-f32 layout for all 16 K-steps ----
    // lane l<16: M=l, K = 4k+{0,1}; lane l>=16: M=l-16, K = 4k+{2,3}
    v2f a[16];
#pragma unroll
    for (int k = 0; k < 16; ++k)
        a[k] = *(const v2f*)&Alds[row * LDA_ + 4 * k + khalf];

    const int tap = row;                              // N across lanes 0..15
    const int ti = tap >> 2, tj = tap & 3;            // (kh, kw)
    const int Y = ti + 2 * ph, X = tj + 2 * pw;
    const int mbase = (lane >> 4) << 3;               // D rows: M=v / M=v+8

    // weight[o, i, p, tap]: taps contiguous (64B per (o,i)); i stride 16384 f
    const float* wbase = w + (size_t)p * TAPS_ + tap;

    for (int c = 0; c < 8; ++c) {
        const int o = wv * 8 + c;
        const float* wb = wbase + (size_t)o * CIN_ * POS_ * TAPS_;

        if (c < 7)  // prefetch next cout's weight panel (global_prefetch_b8)
            __builtin_prefetch(wb + (size_t)CIN_ * POS_ * TAPS_, 0, 1);

        v8f acc = {0.f, 0.f, 0.f, 0.f, 0.f, 0.f, 0.f, 0.f};
#pragma unroll
        for (int k = 0; k < 16; ++k) {
            // B 4x16-f32 layout: lane l<16: N=l, K=4k+{0,1};
            //                    lane l>=16: N=l-16, K=4k+{2,3}
            const int i0 = 4 * k + khalf;
            v2f bm;
            bm.x = wb[(size_t)i0 * (POS_ * TAPS_)];
            bm.y = wb[(size_t)(i0 + 1) * (POS_ * TAPS_)];
            acc = __builtin_amdgcn_wmma_f32_16x16x4_f32(
                false, a[k], false, bm, (short)0, acc, false, false);
        }

        // ---- overlap-add store: D[M=batch][N=tap] -> out[b,o,Y,X] ----
        // VGPR v: lanes 0-15 hold M=v, lanes 16-31 hold M=v+8.
#pragma unroll
        for (int v = 0; v < 8; ++v) {
            int m = v + mbase;
            size_t oi = (((size_t)m * COUT_ + o) * HOUT_ + Y) * HOUT_ + X;
            out[oi] += acc[v];
        }
    }
}

extern "C" void kernel_launch(void* const* d_in, const int* in_sizes, int n_in,
                              void* d_out, int out_size, void* d_ws, size_t ws_size,
                              hipStream_t stream) {
    const float* x    = (const float*)d_in[0];
    const float* wgt  = (const float*)d_in[1];
    const float* bias = (const float*)d_in[2];
    float* out = (float*)d_out;

    const int per = COUT_ * HOUT_ * HOUT_;            // 278784
    bias_broadcast_kernel<<<(out_size + 255) / 256, 256, 0, stream>>>(
        bias, out, out_size, per);

    // 4 parity phases: disjoint output writers within each phase,
    // phases ordered by the stream -> fully deterministic overlap-add.
    for (int par = 0; par < 4; ++par)
        deconv_wmma_kernel<<<256, 256, 0, stream>>>(x, wgt, out, par);
}